// GINN_inputLayer_31215822307668
// MI455X (gfx1250) — compile-verified
//
#include <hip/hip_runtime.h>

// ---------------------------------------------------------------------------
// u2[b,k] = sum_n x[b,k,n] * W[k,n];  out = tanh(u2)
// B=64, K=4096, N=256.  Memory-bound (~0.5 flop/byte): stream x (256MB) once
// via async global->LDS DMA, W (4MB) L2-resident staged per block.
// Reduction on the matrix pipe: per wave, one b + 16-wide k tile;
// D = Xtile * Wtile^T via v_wmma_f32_16x16x4_f32, keep diag(D).
// ---------------------------------------------------------------------------

typedef __attribute__((ext_vector_type(2))) float v2f;
typedef __attribute__((ext_vector_type(4))) float v4f;
typedef __attribute__((ext_vector_type(8))) float v8f;

#define KDIM 4096
#define NDIM 256

#define W_STRIDE 260   // 256+4 pad; 260 % 64 == 4 -> conflict-free B-tile ds reads
#define X_STRIDE 36    // 32+4 pad;   36 % 32 == 4 -> conflict-free A-tile ds reads
#define NCHUNK   32    // n-chunk per async stage (16 rows x 32 floats = 2 KB)
#define XBUF_FLOATS (16 * X_STRIDE)          // 576 floats = 2304 B per buffer
#define XBUF_BYTES  (XBUF_FLOATS * 4)

// one 2KB x-chunk: 4 async b128 ops per lane (lane pattern: 8 lanes = 128B line)
__device__ __forceinline__ void async_stage_chunk(unsigned lds_base,
                                                  unsigned voff_base,
                                                  unsigned long long sbase) {
    #pragma unroll
    for (int i = 0; i < 4; ++i) {
        unsigned la = lds_base + i * (4 * X_STRIDE * 4);   // 4 rows per op
        unsigned va = voff_base + i * (4 * NDIM * 4);
        asm volatile("global_load_async_to_lds_b128 %0, %1, %2"
                     :: "v"(la), "v"(va), "s"(sbase) : "memory");
    }
}

__global__ __launch_bounds__(256) void ginn_bdiag_matvec_tanh(
    const float* __restrict__ x,   // [64, 4096, 256]
    const float* __restrict__ W,   // [4096, 256]
    float* __restrict__ out)       // [64, 4096]
{
    __shared__ float ldsW[16 * W_STRIDE];            // W[k0:k0+16, :], padded
    __shared__ float ldsX[8][2][XBUF_FLOATS];        // per-wave double buffer

    const int tid  = threadIdx.x;
    const int wave = tid >> 5;
    const int lane = tid & 31;
    const int row  = lane & 15;     // matrix row (k index within tile)
    const int half = lane >> 4;     // contraction half (K pairs {0,1} vs {2,3})

    const int ktile = blockIdx.x & 255;   // 256 k tiles
    const int btile = blockIdx.x >> 8;    // 8 b tiles of 8
    const int k0 = ktile * 16;
    const int b  = btile * 8 + wave;

    // ---- async x staging setup (GVS: uniform SGPR base + per-lane voffset) ----
    const float* xbase = x + ((size_t)b * KDIM + k0) * NDIM;
    const unsigned long long sbase = (unsigned long long)(uintptr_t)xbase;
    // per lane: row r = 4*i + (lane>>3), 16B column chunk c = (lane&7)*16 bytes
    const unsigned voff0 = (unsigned)((lane >> 3) * (NDIM * 4) + (lane & 7) * 16);
    const unsigned lds0  = (unsigned)(uintptr_t)&ldsX[wave][0][0]
                         + (unsigned)((lane >> 3) * (X_STRIDE * 4) + (lane & 7) * 16);

    // kick off chunks 0 and 1 (double buffer) before W staging
    async_stage_chunk(lds0,              voff0,                sbase);   // ch0 -> buf0
    async_stage_chunk(lds0 + XBUF_BYTES, voff0 + NCHUNK * 4,   sbase);   // ch1 -> buf1

    // ---- stage W tile cooperatively (coalesced b128, one time) ----
    #pragma unroll
    for (int i = 0; i < 4; ++i) {
        int fi = (i * 256 + tid) * 4;        // flat float idx into 16x256 tile
        int r  = fi >> 8;
        int c  = fi & 255;
        v4f wv = *(const v4f*)(W + (size_t)(k0 + r) * NDIM + c);
        *(v4f*)(&ldsW[r * W_STRIDE + c]) = wv;
    }
    __syncthreads();

    v8f acc = {};

    #pragma unroll
    for (int nb = 0; nb < 8; ++nb) {
        const int buf = nb & 1;
        // chunks nb, nb+1 outstanding (8 ops); <=4 left => chunk nb landed
        if (nb < 7) asm volatile("s_wait_asynccnt 0x4" ::: "memory");
        else        asm volatile("s_wait_asynccnt 0x0" ::: "memory");

        const float* mx = &ldsX[wave][buf][0];
        const int nbase = nb * NCHUNK;
        #pragma unroll
        for (int t = 0; t < 8; ++t) {
            // A[M=row, K=2h+{0,1}] = x[b, k0+row, n+2h+{0,1}]
            v2f a  = *(const v2f*)(&mx[row * X_STRIDE + 4 * t + 2 * half]);
            // B[K=2h+{0,1}, N=row] = W[k0+row, n+2h+{0,1}]
            v2f bm = *(const v2f*)(&ldsW[row * W_STRIDE + nbase + 4 * t + 2 * half]);
            acc = __builtin_amdgcn_wmma_f32_16x16x4_f32(
                      false, a, false, bm, (short)0, acc, false, false);
        }
        // refill this buffer with chunk nb+2 (WAR: drain our ds reads first)
        if (nb < 6) {
            asm volatile("s_wait_dscnt 0x0" ::: "memory");
            async_stage_chunk(lds0 + buf * XBUF_BYTES,
                              voff0 + (nb + 2) * (NCHUNK * 4), sbase);
        }
    }

    // ---- diagonal extract: lane holds D[M,N], M = v + 8*half, N = row ----
    const int v = row & 7;
    float d = acc[0];
    #pragma unroll
    for (int i = 1; i < 8; ++i) d = (v == i) ? acc[i] : d;

    if ((row >> 3) == half) {
        out[(size_t)b * KDIM + k0 + row] = tanhf(d);
    }
}

extern "C" void kernel_launch(void* const* d_in, const int* in_sizes, int n_in,
                              void* d_out, int out_size, void* d_ws, size_t ws_size,
                              hipStream_t stream) {
    const float* x = (const float*)d_in[0];   // [64, 4096, 256]
    const float* W = (const float*)d_in[1];   // [4096, 256]
    float* out = (float*)d_out;               // [64, 4096]

    // 2048 blocks: blockIdx = btile(3b) | ktile(8b); 8 waves/block = 8 b's
    dim3 grid(2048), block(256);
    ginn_bdiag_matvec_tanh<<<grid, block, 0, stream>>>(x, W, out);
}